// Eikonal2D_50560355008954
// MI455X (gfx1250) — compile-verified
//
#include <hip/hip_runtime.h>
#include <stdint.h>

#define GN      256
#define GSTRIDE 258          // 256 + 2 ghost cols/rows
#define BIGV    1.0e6f
#define NITERS  768

typedef float v2f __attribute__((ext_vector_type(2)));
typedef float v8f __attribute__((ext_vector_type(8)));

// ---- Kernel 0: fh = (1/vp) * H, H = 1 ------------------------------------
__global__ void prep_fh(const float* __restrict__ vp, float* __restrict__ fh, int n)
{
    int i = blockIdx.x * blockDim.x + threadIdx.x;
    if (i < n) fh[i] = 1.0f / vp[i];
}

// ---- Kernel 1: persistent single-WGP eikonal solver, grid lives in LDS ---
__global__ __launch_bounds__(1024, 1) void eik_solve(
    const float* __restrict__ fh,
    const int* __restrict__ pix, const int* __restrict__ piy,
    float* __restrict__ ttout)
{
    __shared__ float U[GSTRIDE * GSTRIDE];   // ghosted 258x258 grid (266 KB)
    __shared__ float rowsave[GN];            // snapshot of old row 127

    const int t = threadIdx.x;
    const int c = t & (GN - 1);   // column 0..255 (lanes -> consecutive cols)
    const int q = t >> 8;         // row-segment 0..3
    const int ix = pix[0];
    const int iy = piy[0];

    // init everything (incl. ghosts) to BIG, then plant the source
    for (int i = t; i < GSTRIDE * GSTRIDE; i += 1024) U[i] = BIGV;
    __syncthreads();
    if (t == 0) U[(ix + 1) * GSTRIDE + (iy + 1)] = 0.0f;
    __syncthreads();

    float stage[32];

    // Godunov update for 32 rows of column c, reading only OLD values.
    auto compute32 = [&](int r0, float up) {
        float cur = U[(r0 + 1) * GSTRIDE + (c + 1)];
        #pragma unroll
        for (int i = 0; i < 32; ++i) {
            const int r    = r0 + i;
            const int base = (r + 1) * GSTRIDE + (c + 1);
            float dn  = U[base + GSTRIDE];
            float lf  = U[base - 1];
            float rt  = U[base + 1];
            float fhv = fh[r * GN + c];                 // coalesced, L2-hot
            float a    = fminf(up, dn);
            float b    = fminf(lf, rt);
            float diff = a - b;
            float ufar = fminf(a, b) + fhv;
            float disc = fmaf(diff, -diff, 2.0f * fhv * fhv);
            disc       = fmaxf(disc, 0.0f);
            float unear = 0.5f * (a + b + __builtin_sqrtf(disc));
            float unew  = (fabsf(diff) >= fhv) ? ufar : unear;
            stage[i]    = fminf(cur, unew);
            up = cur; cur = dn;
        }
    };

    auto write32 = [&](int r0) {
        #pragma unroll
        for (int i = 0; i < 32; ++i) {
            const int r = r0 + i;
            float v = stage[i];
            if (r == ix && c == iy) v = 0.0f;           // re-pin source
            U[(r + 1) * GSTRIDE + (c + 1)] = v;
        }
    };

    #pragma unroll 1
    for (int it = 0; it < NITERS; ++it) {
        // ---- top half: rows q*32 .. q*32+31, all reads are old ----
        const int rA = q * 32;
        compute32(rA, U[rA * GSTRIDE + (c + 1)]);       // up-neighbor of r0-1
        float sv = (q == 0) ? U[128 * GSTRIDE + (c + 1)] : 0.0f;  // old row 127
        __syncthreads();
        write32(rA);
        if (q == 0) rowsave[c] = sv;
        __syncthreads();
        // ---- bottom half: rows 128+q*32 ..; row-127 neighbor from snapshot
        const int rB = 128 + q * 32;
        float upB = (q == 0) ? rowsave[c] : U[rB * GSTRIDE + (c + 1)];
        compute32(rB, upB);
        __syncthreads();
        write32(rB);
        __syncthreads();
    }

    // ---- writeback: async LDS -> global (ASYNCcnt path), coalesced ----
    const uint64_t gbase = (uint64_t)(uintptr_t)ttout;
    #pragma unroll 1
    for (int h = 0; h < 2; ++h) {
        const int r0 = h * 128 + q * 32;
        #pragma unroll
        for (int i = 0; i < 32; ++i) {
            const int r = r0 + i;
            unsigned lds = (unsigned)(uintptr_t)&U[(r + 1) * GSTRIDE + (c + 1)];
            int goff = (r * GN + c) * 4;
            asm volatile("global_store_async_from_lds_b32 %0, %1, %2"
                         :: "v"(goff), "v"(lds), "s"(gbase) : "memory");
        }
    }
    asm volatile("s_wait_asynccnt 0x0" ::: "memory");
}

// ---- Kernel 2: bilinear interp + per-block squared-error partial sums ----
__global__ __launch_bounds__(256) void interp_loss(
    const float* __restrict__ tt, const float* __restrict__ loc,
    const float* __restrict__ et, const float* __restrict__ pt,
    float* __restrict__ out, float* __restrict__ partial, int E)
{
    __shared__ float red[256];
    const int tid = threadIdx.x;
    const int e = blockIdx.x * 256 + tid;
    float serr = 0.0f;
    if (e < E) {
        float x = loc[2 * e + 0];
        float y = loc[2 * e + 1];
        float fx0 = fminf(fmaxf(floorf(x), 0.0f), 254.0f);
        float fy0 = fminf(fmaxf(floorf(y), 0.0f), 254.0f);
        int ix0 = (int)fx0, iy0 = (int)fy0;
        float xs = fminf(fmaxf(x, 0.0f), 255.0f);
        float ys = fminf(fmaxf(y, 0.0f), 255.0f);
        float wx0 = xs - fx0, wx1 = (fx0 + 1.0f) - xs;
        float wy0 = ys - fy0, wy1 = (fy0 + 1.0f) - ys;
        const float* p = tt + ix0 * GN + iy0;
        float Q00 = p[0], Q01 = p[1], Q10 = p[GN], Q11 = p[GN + 1];
        float val = Q00 * wx1 * wy1 + Q10 * wx0 * wy1
                  + Q01 * wx1 * wy0 + Q11 * wx0 * wy0;
        float at = et[e] + val;
        out[e] = at;
        float d = at - pt[e];
        serr = d * d;
    }
    red[tid] = serr;
    __syncthreads();
    for (int s = 128; s > 0; s >>= 1) {
        if (tid < s) red[tid] += red[tid + s];
        __syncthreads();
    }
    if (tid == 0) partial[blockIdx.x] = red[0];
}

// ---- Kernel 3: WMMA reduction of partials -> loss ------------------------
// A = ones(16x4), B = 4x16 tile of data, D = A*B + C  => every row of D is
// the column sums of B; chaining C accumulates 64 values per instruction.
__global__ void final_loss(const float* __restrict__ partial, int n,
                           float* __restrict__ lossout, float invE)
{
    const int lane = threadIdx.x;          // single wave32, EXEC all ones
    v2f a; a.x = 1.0f; a.y = 1.0f;         // 16x4 all-ones A operand
    v8f acc = {0.f, 0.f, 0.f, 0.f, 0.f, 0.f, 0.f, 0.f};
    const int nch = (n + 63) >> 6;
    for (int ch = 0; ch < nch; ++ch) {
        int i0 = ch * 64 + lane;
        int i1 = i0 + 32;
        float b0 = partial[(i0 < n) ? i0 : 0]; b0 = (i0 < n) ? b0 : 0.0f;
        float b1 = partial[(i1 < n) ? i1 : 0]; b1 = (i1 < n) ? b1 : 0.0f;
        v2f b; b.x = b0; b.y = b1;
        acc = __builtin_amdgcn_wmma_f32_16x16x4_f32(
            false, a, false, b, (short)0, acc, false, false);
    }
    // c[0]: lanes 0-15 hold D[0][n], lanes 16-31 hold D[8][n]; rows identical
    float s = acc[0];
    #pragma unroll
    for (int off = 16; off > 0; off >>= 1) s += __shfl_xor(s, off, 32);
    s *= 0.5f;                             // counted every column sum twice
    if (lane == 0) *lossout = s * invE;
}

// ---- Host-side launcher ---------------------------------------------------
extern "C" void kernel_launch(void* const* d_in, const int* in_sizes, int n_in,
                              void* d_out, int out_size, void* d_ws, size_t ws_size,
                              hipStream_t stream)
{
    const float* vp  = (const float*)d_in[0];
    const float* loc = (const float*)d_in[1];
    const float* et  = (const float*)d_in[2];
    const float* pt  = (const float*)d_in[3];
    const int*   pix = (const int*)d_in[4];
    const int*   piy = (const int*)d_in[5];
    float* out = (float*)d_out;

    const int NG = GN * GN;
    const int E  = in_sizes[2];
    float* fhbuf   = (float*)d_ws;       // [0, 65536)
    float* ttbuf   = fhbuf + NG;         // [65536, 131072)
    float* partial = ttbuf + NG;         // [131072, 131072+blocks)
    const int blocks = (E + 255) / 256;

    prep_fh   <<<(NG + 255) / 256, 256, 0, stream>>>(vp, fhbuf, NG);
    eik_solve <<<1, 1024, 0, stream>>>(fhbuf, pix, piy, ttbuf);
    interp_loss<<<blocks, 256, 0, stream>>>(ttbuf, loc, et, pt, out, partial, E);
    final_loss<<<1, 32, 0, stream>>>(partial, blocks, out + E, 1.0f / (float)E);
}